// GCNIINodeRegressor_68659347194259
// MI455X (gfx1250) — compile-verified
//
#include <hip/hip_runtime.h>
#include <math.h>

#define N_NODES 100000
#define E_EDGES 1600000
#define IN_DIM  256
#define HID     128
#define LAYERS  8
#define ALPHA   0.1f

typedef float v2f __attribute__((ext_vector_type(2)));
typedef float v8f __attribute__((ext_vector_type(8)));

// ---------------------------------------------------------------- degree prep
__global__ void __launch_bounds__(256) k_deg_init(float* __restrict__ deg, int n) {
    int i = blockIdx.x * blockDim.x + threadIdx.x;
    if (i < n) deg[i] = 1.0f;   // self-loop
}

__global__ void __launch_bounds__(256) k_deg_count(const long long* __restrict__ col,
                                                   float* __restrict__ deg, int e) {
    int i = blockIdx.x * blockDim.x + threadIdx.x;
    if (i < e) atomicAdd(&deg[(int)col[i]], 1.0f);
}

__global__ void __launch_bounds__(256) k_dinv(float* __restrict__ deg, int n) {
    int i = blockIdx.x * blockDim.x + threadIdx.x;
    if (i < n) deg[i] = rsqrtf(deg[i]);
}

// ------------------------------------------------- input GEMM: h0 = relu(x@W_in + b)
// grid.x = N/16 row-tiles; 256 threads = 8 waves; wave w owns cols [16w,16w+16)
__global__ void __launch_bounds__(256) k_gemm_in(const float* __restrict__ X,
                                                 const float* __restrict__ W,
                                                 const float* __restrict__ bias,
                                                 float* __restrict__ H0) {
    const int lane = threadIdx.x & 31;
    const int wave = threadIdx.x >> 5;
    const int row0 = blockIdx.x * 16;
    const int col0 = wave * 16;
    const int r    = lane & 15;     // M (A) or N (B) within tile
    const int hi   = lane >> 4;     // K half-select (0 -> K+0/1, 1 -> K+2/3)

    v8f acc = {};
    const float* __restrict__ xrow = X + (size_t)(row0 + r) * IN_DIM;
    const float* __restrict__ wcol = W + col0 + r;

    #pragma unroll 4
    for (int k = 0; k < IN_DIM; k += 4) {
        v2f a, b;
        a.x = xrow[k + 2 * hi];
        a.y = xrow[k + 2 * hi + 1];
        b.x = wcol[(size_t)(k + 2 * hi)     * HID];
        b.y = wcol[(size_t)(k + 2 * hi + 1) * HID];
        acc = __builtin_amdgcn_wmma_f32_16x16x4_f32(false, a, false, b,
                                                    (short)0, acc, false, false);
    }
    // D layout: VGPR v -> row (v + 8*hi), col = lane&15
    const int c  = col0 + r;
    const float bv = bias[c];
    #pragma unroll
    for (int v = 0; v < 8; ++v) {
        int rr = row0 + v + 8 * hi;
        H0[(size_t)rr * HID + c] = fmaxf(acc[v] + bv, 0.0f);
    }
}

// ------------------------------- s = ALPHA*h0 + (1-ALPHA)*dinv^2*h  (self-loop term)
__global__ void __launch_bounds__(256) k_prop_init(const float* __restrict__ h0,
                                                   const float* __restrict__ h,
                                                   const float* __restrict__ dinv,
                                                   float* __restrict__ s, int total) {
    int i = blockIdx.x * blockDim.x + threadIdx.x;
    if (i < total) {
        float d = dinv[i >> 7];     // node = i / HID
        s[i] = ALPHA * h0[i] + (1.0f - ALPHA) * d * d * h[i];
    }
}

// ----------------------- wave-per-edge gather/scale/scatter: s[dst] += w * h[src]
__global__ void __launch_bounds__(256) k_prop_edges(const long long* __restrict__ rowI,
                                                    const long long* __restrict__ colI,
                                                    const float* __restrict__ dinv,
                                                    const float* __restrict__ h,
                                                    float* __restrict__ s, int e) {
    int wid  = (int)((blockIdx.x * (unsigned)blockDim.x + threadIdx.x) >> 5);
    int lane = threadIdx.x & 31;
    if (wid >= e) return;
    int src = (int)rowI[wid];
    int dst = (int)colI[wid];
    float w = (1.0f - ALPHA) * dinv[src] * dinv[dst];
    const float* __restrict__ hs = h + (size_t)src * HID;
    float* __restrict__ sd = s + (size_t)dst * HID;
    #pragma unroll
    for (int j = 0; j < 4; ++j) {
        int f = lane + 32 * j;                 // unit-stride per atomic instruction
        atomicAdd(&sd[f], w * hs[f]);
    }
}

// ------------------ layer GEMM: h = relu((1-beta)*s + beta*(s@Wl)), WMMA f32 K=128
__global__ void __launch_bounds__(256) k_gemm_layer(const float* __restrict__ S,
                                                    const float* __restrict__ W,
                                                    float beta,
                                                    float* __restrict__ H) {
    const int lane = threadIdx.x & 31;
    const int wave = threadIdx.x >> 5;
    const int row0 = blockIdx.x * 16;
    const int col0 = wave * 16;
    const int r    = lane & 15;
    const int hi   = lane >> 4;

    v8f acc = {};
    const float* __restrict__ srow = S + (size_t)(row0 + r) * HID;
    const float* __restrict__ wcol = W + col0 + r;

    #pragma unroll 4
    for (int k = 0; k < HID; k += 4) {
        v2f a, b;
        a.x = srow[k + 2 * hi];
        a.y = srow[k + 2 * hi + 1];
        b.x = wcol[(size_t)(k + 2 * hi)     * HID];
        b.y = wcol[(size_t)(k + 2 * hi + 1) * HID];
        acc = __builtin_amdgcn_wmma_f32_16x16x4_f32(false, a, false, b,
                                                    (short)0, acc, false, false);
    }
    const int c = col0 + r;
    const float omb = 1.0f - beta;
    #pragma unroll
    for (int v = 0; v < 8; ++v) {
        int rr = row0 + v + 8 * hi;
        float sv = S[(size_t)rr * HID + c];
        H[(size_t)rr * HID + c] = fmaxf(omb * sv + beta * acc[v], 0.0f);
    }
}

// -------------------------- readout: out = h @ W_out + b_out (wave-per-node dot128)
__global__ void __launch_bounds__(256) k_out(const float* __restrict__ h,
                                             const float* __restrict__ Wout,
                                             const float* __restrict__ bout,
                                             float* __restrict__ out, int n) {
    int wid  = (int)((blockIdx.x * (unsigned)blockDim.x + threadIdx.x) >> 5);
    int lane = threadIdx.x & 31;
    if (wid >= n) return;
    const float* __restrict__ hr = h + (size_t)wid * HID;
    float acc = 0.0f;
    #pragma unroll
    for (int j = 0; j < 4; ++j) acc += hr[lane + 32 * j] * Wout[lane + 32 * j];
    #pragma unroll
    for (int off = 16; off > 0; off >>= 1) acc += __shfl_xor(acc, off, 32);
    if (lane == 0) out[wid] = acc + bout[0];
}

// ----------------------------------------------------------------------------
extern "C" void kernel_launch(void* const* d_in, const int* in_sizes, int n_in,
                              void* d_out, int out_size, void* d_ws, size_t ws_size,
                              hipStream_t stream) {
    const float*     x      = (const float*)d_in[0];
    const long long* ei     = (const long long*)d_in[1];
    const float*     W_in   = (const float*)d_in[2];
    const float*     b_in   = (const float*)d_in[3];
    const float*     convsW = (const float*)d_in[4];
    const float*     W_out  = (const float*)d_in[5];
    const float*     b_out  = (const float*)d_in[6];
    float*           out    = (float*)d_out;

    // workspace carve (all f32): dinv[N] | h0[N*HID] | h[N*HID] | s[N*HID]
    float* dinv = (float*)d_ws;
    float* h0   = dinv + N_NODES;
    float* h    = h0 + (size_t)N_NODES * HID;
    float* s    = h  + (size_t)N_NODES * HID;

    const long long* rowI = ei;            // edge_index[0] : sources
    const long long* colI = ei + E_EDGES;  // edge_index[1] : destinations

    const int T = 256;
    k_deg_init <<<(N_NODES + T - 1) / T, T, 0, stream>>>(dinv, N_NODES);
    k_deg_count<<<(E_EDGES + T - 1) / T, T, 0, stream>>>(colI, dinv, E_EDGES);
    k_dinv     <<<(N_NODES + T - 1) / T, T, 0, stream>>>(dinv, N_NODES);

    k_gemm_in<<<N_NODES / 16, T, 0, stream>>>(x, W_in, b_in, h0);

    const int totalF = N_NODES * HID;
    for (int i = 0; i < LAYERS; ++i) {
        const float* hin = (i == 0) ? h0 : h;   // layer 0 reads h0 directly
        k_prop_init <<<(totalF + T - 1) / T, T, 0, stream>>>(h0, hin, dinv, s, totalF);
        k_prop_edges<<<E_EDGES / 8, T, 0, stream>>>(rowI, colI, dinv, hin, s, E_EDGES);
        float beta = logf(0.5f / (float)(i + 1) + 1.0f);
        k_gemm_layer<<<N_NODES / 16, T, 0, stream>>>(s, convsW + (size_t)i * HID * HID,
                                                     beta, h);
    }

    k_out<<<(N_NODES * 32 + T - 1) / T, T, 0, stream>>>(h, W_out, b_out, out, N_NODES);
}